// MNB_24111946400019
// MI455X (gfx1250) — compile-verified
//
#include <hip/hip_runtime.h>

#define SEQ 200
#define BATCH 4096
#define VOCAB 50000
#define DOCS_PER_BLOCK 8
#define BLOCK 256
#define BITMAP_DW 1568   // >= ceil(50000/32)=1563, padded to multiple of 32

typedef __attribute__((address_space(1))) int GInt;   // global int
typedef __attribute__((address_space(3))) int LInt;   // LDS int

__global__ __launch_bounds__(BLOCK)
void MNB_presence_dot_kernel(const int* __restrict__ text,
                             const float* __restrict__ w,
                             const float* __restrict__ bias,
                             float* __restrict__ out)
{
    __shared__ unsigned bitmap[DOCS_PER_BLOCK * BITMAP_DW];   // 50176 B
    __shared__ int      textTile[DOCS_PER_BLOCK * SEQ];       //  6400 B

    const int tid     = threadIdx.x;
    const int docBase = blockIdx.x * DOCS_PER_BLOCK;

    // ---- 1) zero the per-doc presence bit-sets (ds_store_b128) ----
    uint4* bm4 = (uint4*)bitmap;
    const int n4 = (DOCS_PER_BLOCK * BITMAP_DW) / 4;          // 3136
    for (int i = tid; i < n4; i += BLOCK)
        bm4[i] = make_uint4(0u, 0u, 0u, 0u);

    // ---- 2) stage text tile [8 docs x 200 tokens] into LDS ----
    // global layout: text[t*BATCH + b]; LDS layout: textTile[d*SEQ + t]
    const int* gsrc = text + docBase;
    for (int e = tid; e < DOCS_PER_BLOCK * SEQ; e += BLOCK) {
        const int d = e & (DOCS_PER_BLOCK - 1);
        const int t = e >> 3;
#if __has_builtin(__builtin_amdgcn_global_load_async_to_lds_b32)
        __builtin_amdgcn_global_load_async_to_lds_b32(
            (GInt*)(gsrc + (size_t)t * BATCH + d),
            (LInt*)&textTile[d * SEQ + t],
            /*imm offset*/0, /*cpol*/0);
#else
        textTile[d * SEQ + t] = gsrc[(size_t)t * BATCH + d];
#endif
    }
#if __has_builtin(__builtin_amdgcn_global_load_async_to_lds_b32)
#  if __has_builtin(__builtin_amdgcn_s_wait_asynccnt)
    __builtin_amdgcn_s_wait_asynccnt(0);
#  else
    asm volatile("s_wait_asynccnt 0x0" ::: "memory");
#  endif
#endif
    __syncthreads();

    // ---- 3) one wave32 per document: bitmap dedup + weight gather ----
    const int lane = tid & 31;
    const int wav  = tid >> 5;                 // doc within block
    unsigned*  myBm  = bitmap   + wav * BITMAP_DW;
    const int* myTok = textTile + wav * SEQ;

    float sum = 0.0f;
    #pragma unroll
    for (int t = lane; t < SEQ; t += 32) {
        const unsigned tok = (unsigned)myTok[t];
        const unsigned dw  = tok >> 5;
        const unsigned bit = 1u << (tok & 31u);
        const unsigned old = atomicOr(&myBm[dw], bit);   // ds_or_rtn_b32
        if ((old & bit) == 0u)                            // first occurrence only
            sum += w[tok];
    }

    // ---- 4) wave32 reduction ----
    #pragma unroll
    for (int off = 16; off > 0; off >>= 1)
        sum += __shfl_xor(sum, off, 32);

    if (lane == 0)
        out[docBase + wav] = sum + bias[0];
}

extern "C" void kernel_launch(void* const* d_in, const int* in_sizes, int n_in,
                              void* d_out, int out_size, void* d_ws, size_t ws_size,
                              hipStream_t stream) {
    const int*   text = (const int*)  d_in[0];   // [SEQ, BATCH] int32 token ids
    const float* w    = (const float*)d_in[1];   // [1, V] float32
    const float* bias = (const float*)d_in[2];   // [1] float32
    float*       out  = (float*)d_out;           // [BATCH, 1] float32

    const int grid = BATCH / DOCS_PER_BLOCK;     // 512 blocks x 256 threads
    MNB_presence_dot_kernel<<<grid, BLOCK, 0, stream>>>(text, w, bias, out);
}